// Model_56538949485253
// MI455X (gfx1250) — compile-verified
//
#include <hip/hip_runtime.h>

typedef float v2f __attribute__((ext_vector_type(2)));
typedef float v8f __attribute__((ext_vector_type(8)));

constexpr int N_  = 100000;   // nodes per type
constexpr int E_  = 2000000;  // edges per direction
constexpr int EL_ = 500000;   // label edges
constexpr int D_  = 128;      // feature dim

// ---------------------------------------------------------------- utilities

__global__ void zero_kernel(float* __restrict__ p, long n) {
    long i = (long)blockIdx.x * blockDim.x + threadIdx.x;
    if (i < n) p[i] = 0.0f;
}

__device__ __forceinline__ void atomic_add_f32_dev(float* p, float v) {
    // Native L2 fp32 atomic, device scope (many WGPs RMW the same rows).
    // Tracked with STOREcnt; s_endpgm's implicit wait-idle covers completion.
    asm volatile("global_atomic_add_f32 %0, %1, off scope:SCOPE_DEV"
                 :: "v"(p), "v"(v) : "memory");
}

// One wave32 per edge: lane handles 4 consecutive floats of the feature row.
// agg is 51MB -> resident in the 192MB L2, where the atomics execute.
__global__ void scatter_kernel(const float* __restrict__ xsrc,
                               const int* __restrict__ edges,
                               float* __restrict__ agg,
                               float* __restrict__ cnt) {
    int gid  = blockIdx.x * blockDim.x + threadIdx.x;
    int e    = gid >> 5;
    int lane = threadIdx.x & 31;
    if (e >= E_) return;
    int s = edges[e];        // src row
    int d = edges[E_ + e];   // dst row
    float4 v = *(const float4*)(xsrc + (size_t)s * D_ + lane * 4);
    float* dst = agg + (size_t)d * D_ + lane * 4;
    atomic_add_f32_dev(dst + 0, v.x);
    atomic_add_f32_dev(dst + 1, v.y);
    atomic_add_f32_dev(dst + 2, v.z);
    atomic_add_f32_dev(dst + 3, v.w);
    if (lane == 0) atomic_add_f32_dev(cnt + d, 1.0f);
}

// ---------------------------------------------------------------- fused SAGE linear
// out[n,o] = sum_k (agg[n,k]/max(cnt,1))*W1[o,k] + sum_k A2[n,k]*W2[o,k] + bias[o]
// Block = 256 threads = 8 waves covering a 16-row tile; wave w owns columns
// [16w,16w+16). Mean division is fused into the LDS staging of the A1 tile;
// both A tiles are staged once per block (kills the 8x redundant global reads).
// K-loop of V_WMMA_F32_16X16X4_F32; fragment layout per ISA 7.12.2:
//   lanes 0-15 : K = kk+0 (v0), kk+1 (v1);  lanes 16-31 : K = kk+2, kk+3
__global__ void sage_gemm(const float* __restrict__ AGG, const float* __restrict__ A2,
                          const float* __restrict__ cnt,
                          const float* __restrict__ W1, const float* __restrict__ W2,
                          const float* __restrict__ bias, float* __restrict__ out,
                          int relu) {
    __shared__ float a1t[16][132];   // mean tile   (pad 128->132: no bank conflicts)
    __shared__ float a2t[16][132];   // x_dst tile
    int mbase = blockIdx.x * 16;

    // Phase 1: stage A tiles; fold mean division into staging.
    {
        int row = threadIdx.x >> 4;   // tile row
        int seg = threadIdx.x & 15;   // 8-float segment
        int n   = mbase + row;
        float inv = 1.0f / fmaxf(cnt[n], 1.0f);
        const float* p1 = AGG + (size_t)n * D_ + seg * 8;
        const float* p2 = A2  + (size_t)n * D_ + seg * 8;
        float4 s0 = *(const float4*)(p1);
        float4 s1 = *(const float4*)(p1 + 4);
        float4 d0 = *(const float4*)(p2);
        float4 d1 = *(const float4*)(p2 + 4);
        float* q1 = &a1t[row][seg * 8];
        float* q2 = &a2t[row][seg * 8];
        q1[0] = s0.x * inv; q1[1] = s0.y * inv; q1[2] = s0.z * inv; q1[3] = s0.w * inv;
        q1[4] = s1.x * inv; q1[5] = s1.y * inv; q1[6] = s1.z * inv; q1[7] = s1.w * inv;
        q2[0] = d0.x; q2[1] = d0.y; q2[2] = d0.z; q2[3] = d0.w;
        q2[4] = d1.x; q2[5] = d1.y; q2[6] = d1.z; q2[7] = d1.w;
    }
    __syncthreads();

    int lane = threadIdx.x & 31;
    int wave = threadIdx.x >> 5;
    int half = lane >> 4;
    int lm   = lane & 15;
    int col  = wave * 16 + lm;       // output column (o), same for all 8 accs

    const float* w1p = W1 + (size_t)col * D_ + half * 2;
    const float* w2p = W2 + (size_t)col * D_ + half * 2;

    v8f acc = {};
#pragma unroll 8
    for (int kk = 0; kk < D_; kk += 4) {
        v2f a = *(const v2f*)(&a1t[lm][kk + half * 2]);
        v2f b = *(const v2f*)(w1p + kk);
        acc = __builtin_amdgcn_wmma_f32_16x16x4_f32(false, a, false, b,
                                                    (short)0, acc, false, false);
    }
#pragma unroll 8
    for (int kk = 0; kk < D_; kk += 4) {
        v2f a = *(const v2f*)(&a2t[lm][kk + half * 2]);
        v2f b = *(const v2f*)(w2p + kk);
        acc = __builtin_amdgcn_wmma_f32_16x16x4_f32(false, a, false, b,
                                                    (short)0, acc, false, false);
    }

    float bv = bias[col];
    float* op = out + (size_t)(mbase + half * 8) * D_ + wave * 16 + lm;
#pragma unroll
    for (int r = 0; r < 8; ++r) {
        float v = acc[r] + bv;
        if (relu) v = fmaxf(v, 0.0f);
        op[(size_t)r * D_] = v;      // row = mbase + r + half*8
    }
}

// ---------------------------------------------------------------- fused decoder
// Per block: 16 label edges. Stage z = z_c[i0]*z_r[i1] (16x128) in LDS once,
// h = relu(z @ W1^T + b1) via WMMA, out = h . w2 + b2 via in-wave shuffle +
// cross-wave LDS reduction. Avoids materializing the 256MB z tensor.
__global__ void decoder_kernel(const float* __restrict__ zc, const float* __restrict__ zr,
                               const int* __restrict__ eli,
                               const float* __restrict__ W1, const float* __restrict__ b1,
                               const float* __restrict__ w2, const float* __restrict__ b2,
                               float* __restrict__ out) {
    __shared__ float zt[16][132];
    __shared__ float part[8][16];
    int ebase = blockIdx.x * 16;

    // Phase 1: gather + elementwise product into LDS
    {
        int row = threadIdx.x >> 4;
        int seg = threadIdx.x & 15;
        int e  = ebase + row;
        int i0 = eli[e];              // index into z_c
        int i1 = eli[EL_ + e];        // index into z_r
        const float* pc = zc + (size_t)i0 * D_ + seg * 8;
        const float* pr = zr + (size_t)i1 * D_ + seg * 8;
        float4 c0 = *(const float4*)(pc);
        float4 c1 = *(const float4*)(pc + 4);
        float4 r0 = *(const float4*)(pr);
        float4 r1 = *(const float4*)(pr + 4);
        float* zp = &zt[row][seg * 8];
        zp[0] = c0.x * r0.x; zp[1] = c0.y * r0.y; zp[2] = c0.z * r0.z; zp[3] = c0.w * r0.w;
        zp[4] = c1.x * r1.x; zp[5] = c1.y * r1.y; zp[6] = c1.z * r1.z; zp[7] = c1.w * r1.w;
    }
    __syncthreads();

    int lane = threadIdx.x & 31;
    int wave = threadIdx.x >> 5;
    int half = lane >> 4;
    int lm   = lane & 15;
    int col  = wave * 16 + lm;

    v8f acc = {};
    const float* wp = W1 + (size_t)col * D_ + half * 2;
#pragma unroll 8
    for (int kk = 0; kk < D_; kk += 4) {
        v2f a = *(const v2f*)(&zt[lm][kk + half * 2]);
        v2f b = *(const v2f*)(wp + kk);
        acc = __builtin_amdgcn_wmma_f32_16x16x4_f32(false, a, false, b,
                                                    (short)0, acc, false, false);
    }

    float bv = b1[col];
    float wv = w2[col];
#pragma unroll
    for (int r = 0; r < 8; ++r) {
        float p = fmaxf(acc[r] + bv, 0.0f) * wv;
        p += __shfl_xor(p, 1, 32);    // masks 1..8 stay within each 16-lane half
        p += __shfl_xor(p, 2, 32);
        p += __shfl_xor(p, 4, 32);
        p += __shfl_xor(p, 8, 32);
        if (lm == 0) part[wave][half * 8 + r] = p;
    }
    __syncthreads();

    if (threadIdx.x < 16) {
        float s = b2[0];
#pragma unroll
        for (int w = 0; w < 8; ++w) s += part[w][threadIdx.x];
        out[ebase + threadIdx.x] = s;
    }
}

// ---------------------------------------------------------------- launch

extern "C" void kernel_launch(void* const* d_in, const int* in_sizes, int n_in,
                              void* d_out, int out_size, void* d_ws, size_t ws_size,
                              hipStream_t stream) {
    (void)in_sizes; (void)n_in; (void)out_size; (void)ws_size;

    const float* x_c      = (const float*)d_in[0];
    const float* x_r      = (const float*)d_in[1];
    const int*   e_c2r    = (const int*)d_in[2];
    const int*   e_r2c    = (const int*)d_in[3];
    const int*   eli      = (const int*)d_in[4];
    const float* W1_c2r_l = (const float*)d_in[5];
    const float* W1_c2r_r = (const float*)d_in[6];
    const float* W1_r2c_l = (const float*)d_in[7];
    const float* W1_r2c_r = (const float*)d_in[8];
    const float* W2_c2r_l = (const float*)d_in[9];
    const float* W2_c2r_r = (const float*)d_in[10];
    const float* W2_r2c_l = (const float*)d_in[11];
    const float* W2_r2c_r = (const float*)d_in[12];
    const float* b1_c2r   = (const float*)d_in[13];
    const float* b1_r2c   = (const float*)d_in[14];
    const float* b2_c2r   = (const float*)d_in[15];
    const float* b2_r2c   = (const float*)d_in[16];
    const float* dec_W1   = (const float*)d_in[17];
    const float* dec_b1   = (const float*)d_in[18];
    const float* dec_W2   = (const float*)d_in[19];
    const float* dec_b2   = (const float*)d_in[20];
    float* out = (float*)d_out;

    const size_t ND = (size_t)N_ * D_;
    float* agg = (float*)d_ws;     // [N,D] edge-sum (mean fused into GEMM staging)
    float* cnt = agg + ND;         // [N]   (contiguous with agg for one zero-fill)
    float* h_r = cnt + N_;         // [N,D]
    float* h_c = h_r + ND;         // [N,D]
    float* z_r = h_c + ND;         // [N,D]
    float* z_c = z_r + ND;         // [N,D]

    auto conv = [&](const float* xsrc, const float* xdst, const int* edges,
                    const float* Wl, const float* Wr, const float* b,
                    float* dst, int relu) {
        long zn = (long)(ND + N_);
        zero_kernel<<<(unsigned)((zn + 255) / 256), 256, 0, stream>>>(agg, zn);
        scatter_kernel<<<E_ / 8, 256, 0, stream>>>(xsrc, edges, agg, cnt);
        sage_gemm<<<N_ / 16, 256, 0, stream>>>(agg, xdst, cnt, Wl, Wr, b, dst, relu);
    };

    // layer 1 (relu)
    conv(x_c, x_r, e_c2r, W1_c2r_l, W1_c2r_r, b1_c2r, h_r, 1);
    conv(x_r, x_c, e_r2c, W1_r2c_l, W1_r2c_r, b1_r2c, h_c, 1);
    // layer 2 (no relu)
    conv(h_c, h_r, e_c2r, W2_c2r_l, W2_c2r_r, b2_c2r, z_r, 0);
    conv(h_r, h_c, e_r2c, W2_r2c_l, W2_r2c_r, b2_r2c, z_c, 0);
    // decoder
    decoder_kernel<<<EL_ / 16, 256, 0, stream>>>(z_c, z_r, eli, dec_W1, dec_b1,
                                                 dec_W2, dec_b2, out);
}